// VectorQuantizerEMA_19146964206408
// MI455X (gfx1250) — compile-verified
//
#include <hip/hip_runtime.h>
#include <hip/hip_bf16.h>

typedef float v2f __attribute__((ext_vector_type(2)));
typedef float v8f __attribute__((ext_vector_type(8)));

#define NUM_EMB 1024
#define EMB_DIM 64
#define N_FLAT  32768            // 32*32*32 flattened vectors
#define DECAY 0.99f
#define OMD   0.01f              // 1 - DECAY
#define EPS   1e-5f
#define COMMIT 0.25f

// ---- d_out layout (float offsets), tuple return order ----
#define OUT_Q      0                         // [32,64,32,32] = 2097152
#define OUT_LOSS   2097152                   // scalar
#define OUT_ENC    2097153                   // [32768,1024] one-hot
#define OUT_NEWEMB (2097153 + 33554432)      // [1024,64]
#define OUT_NCS    (OUT_NEWEMB + 65536)      // [1024]
#define OUT_NEMAW  (OUT_NCS + 1024)          // [1024,64]

// ---- d_ws layout (float offsets) ----
#define WS_COUNTS 0                          // [1024] cluster counts
#define WS_DW     1024                       // [1024,64] dw accumulator
#define WS_LOSS   (1024 + 65536)             // scalar SSE accumulator
#define WS_CNORM  (WS_LOSS + 1)              // [1024] ||e_k||^2
#define WS_ZERO_N (WS_LOSS + 1)              // floats zeroed each launch
#define WS_TOTAL  (WS_CNORM + 1024)

// ---------------------------------------------------------------------------
// Kernel 0: zero accumulators (must happen every launch) + code norms
// ---------------------------------------------------------------------------
__global__ void vq_init(const float* __restrict__ emb, float* __restrict__ ws) {
    int i = blockIdx.x * blockDim.x + threadIdx.x;
    if (i < WS_ZERO_N) ws[i] = 0.0f;
    if (i < NUM_EMB) {
        const float* e = emb + i * EMB_DIM;
        float s = 0.0f;
        #pragma unroll
        for (int d = 0; d < EMB_DIM; ++d) { float v = e[d]; s += v * v; }
        ws[WS_CNORM + i] = s;
    }
}

// ---------------------------------------------------------------------------
// Kernel 1: WMMA distances + argmin + all per-row outputs.
// 128 threads (4 wave32), one 16-row tile per workgroup, 2048 workgroups.
// ---------------------------------------------------------------------------
#define LDA 66  // padded LDS row stride (floats): no 64-bank conflicts, 8B aligned

__global__ void __launch_bounds__(128)
vq_main(const float* __restrict__ in,   // NCHW [32,64,32,32]
        const float* __restrict__ emb,  // [1024,64]
        float* __restrict__ out,
        float* __restrict__ ws) {
    __shared__ float As[16 * LDA];
    __shared__ float wMinV[4][16];
    __shared__ int   wMinI[4][16];
    __shared__ int   encIdx[16];
    __shared__ float lossLds;

    const int tid  = threadIdx.x;
    const int lane = tid & 31;
    const int wave = tid >> 5;
    const int rowBase = blockIdx.x * 16;   // all 16 rows share one batch image b
    const int bImg = rowBase >> 10;
    const int hw0  = rowBase & 1023;

    if (tid == 0) lossLds = 0.0f;

    // ---- Phase A: gather 16 NHWC-flat rows into LDS (coalesced over hw) ----
    // flat[r][d] = in[b*65536 + d*1024 + hw],  r = b*1024 + hw
    for (int i = tid; i < 16 * EMB_DIM; i += 128) {
        int m = i & 15;
        int d = i >> 4;
        As[m * LDA + d] = in[bImg * 65536 + d * 1024 + hw0 + m];
    }
    __syncthreads();

    // ---- Phase B: per-wave column sweep with V_WMMA_F32_16X16X4_F32 ----
    // argmin(||x||^2 + ||e||^2 - 2 x.e) == argmin(||e||^2 - 2 x.e)
    const int hv  = lane >> 4;   // half: 0 -> rows j, 1 -> rows j+8 (C/D layout)
    const int l15 = lane & 15;

    // A fragments are ct-invariant: hoist all 16 (K=4 each) out of the loop.
    v2f afrag[16];
    #pragma unroll
    for (int kk = 0; kk < 16; ++kk)
        afrag[kk] = *(const v2f*)&As[l15 * LDA + kk * 4 + hv * 2];

    float runV[8];
    int   runI[8];
    #pragma unroll
    for (int j = 0; j < 8; ++j) { runV[j] = 3.0e38f; runI[j] = 0; }

    for (int ct = wave; ct < NUM_EMB / 16; ct += 4) {
        const int col = ct * 16 + l15;
        const float* ecol = emb + col * EMB_DIM;   // embedding row == B column
        // Preload ALL B fragments so the 16 b64 loads pipeline as one clause,
        // then run the 16 WMMAs back-to-back (single loadcnt wait per tile).
        v2f bfrag[16];
        #pragma unroll
        for (int kk = 0; kk < 16; ++kk)
            bfrag[kk] = *(const v2f*)(ecol + kk * 4 + hv * 2);
        float cn = ws[WS_CNORM + col];

        v8f c = {};
        #pragma unroll
        for (int kk = 0; kk < 16; ++kk)
            c = __builtin_amdgcn_wmma_f32_16x16x4_f32(
                    false, afrag[kk], false, bfrag[kk], (short)0, c, false, false);

        // Branchless per-lane running min (per-lane cols strictly increase, so
        // '<' keeps the first/lowest index automatically).
        #pragma unroll
        for (int j = 0; j < 8; ++j) {
            float v = __builtin_fmaf(-2.0f, c[j], cn);   // row (j + 8*hv)
            bool lt = v < runV[j];
            runV[j] = lt ? v : runV[j];
            runI[j] = lt ? col : runI[j];
        }
    }

    // Cross-lane argmin ONCE per row (width-16 halves), with index tie-break.
    #pragma unroll
    for (int j = 0; j < 8; ++j) {
        float v = runV[j]; int idx = runI[j];
        #pragma unroll
        for (int off = 8; off >= 1; off >>= 1) {
            float ov = __shfl_xor(v, off, 16);
            int   oi = __shfl_xor(idx, off, 16);
            if (ov < v || (ov == v && oi < idx)) { v = ov; idx = oi; }
        }
        runV[j] = v; runI[j] = idx;
    }
    if (l15 == 0) {
        #pragma unroll
        for (int j = 0; j < 8; ++j) {
            wMinV[wave][hv * 8 + j] = runV[j];
            wMinI[wave][hv * 8 + j] = runI[j];
        }
    }
    __syncthreads();

    // ---- combine across the 4 waves, commit histogram ----
    if (tid < 16) {
        float v = wMinV[0][tid]; int idx = wMinI[0][tid];
        #pragma unroll
        for (int w = 1; w < 4; ++w) {
            float ov = wMinV[w][tid]; int oi = wMinI[w][tid];
            if (ov < v || (ov == v && oi < idx)) { v = ov; idx = oi; }
        }
        encIdx[tid] = idx;
        atomicAdd(&ws[WS_COUNTS + idx], 1.0f);     // encodings_sum
    }
    __syncthreads();

    // ---- Phase C1: loss partial + dw scatter (dw = E^T x) ----
    float lossAcc = 0.0f;
    for (int i = tid; i < 16 * EMB_DIM; i += 128) {
        int m = i & 15;
        int d = i >> 4;
        int e = encIdx[m];
        float q = emb[e * EMB_DIM + d];
        float x = As[m * LDA + d];
        float diff = q - x;
        lossAcc += diff * diff;
        atomicAdd(&ws[WS_DW + e * EMB_DIM + d], x);
    }
    atomicAdd(&lossLds, lossAcc);

    // ---- Phase C2: quantized back to NCHW (STE value == quantized), b128 ----
    for (int u = tid; u < 256; u += 128) {          // 64 d * 4 groups of 4 rows
        int d  = u >> 2;
        int mg = (u & 3) * 4;
        float4 q;
        q.x = emb[encIdx[mg + 0] * EMB_DIM + d];
        q.y = emb[encIdx[mg + 1] * EMB_DIM + d];
        q.z = emb[encIdx[mg + 2] * EMB_DIM + d];
        q.w = emb[encIdx[mg + 3] * EMB_DIM + d];
        *(float4*)(out + OUT_Q + bImg * 65536 + d * 1024 + hw0 + mg) = q;
    }

    // ---- Phase C3: one-hot encodings, the 134 MB stream ----
    // Base OUT_ENC is ==1 (mod 4) floats, so k = 4s+3 gives 16B-aligned b128
    // stores covering k=3..1022; slot s==255 mops up k={1023,0,1,2} scalar.
    for (int i = tid; i < 16 * 256; i += 128) {
        int m  = i >> 8;
        int s  = i & 255;
        int e  = encIdx[m];
        float* rowp = out + OUT_ENC + (rowBase + m) * NUM_EMB;
        if (s < 255) {
            int k0 = s * 4 + 3;
            float4 v;
            v.x = (k0 + 0 == e) ? 1.0f : 0.0f;
            v.y = (k0 + 1 == e) ? 1.0f : 0.0f;
            v.z = (k0 + 2 == e) ? 1.0f : 0.0f;
            v.w = (k0 + 3 == e) ? 1.0f : 0.0f;
            *(float4*)(rowp + k0) = v;
        } else {
            rowp[1023] = (1023 == e) ? 1.0f : 0.0f;
            rowp[0]    = (0 == e) ? 1.0f : 0.0f;
            rowp[1]    = (1 == e) ? 1.0f : 0.0f;
            rowp[2]    = (2 == e) ? 1.0f : 0.0f;
        }
    }
    __syncthreads();
    if (tid == 0) atomicAdd(&ws[WS_LOSS], lossLds);
}

// ---------------------------------------------------------------------------
// Kernel 2: EMA updates, Laplace smoothing, normalized codebook, loss scalar
// ---------------------------------------------------------------------------
__global__ void __launch_bounds__(1024)
vq_finalize(const float* __restrict__ ema_cs, const float* __restrict__ ema_w,
            float* __restrict__ out, const float* __restrict__ ws) {
    __shared__ float red[32];
    const int k = threadIdx.x;

    float ncs = ema_cs[k] * DECAY + OMD * ws[WS_COUNTS + k];
    out[OUT_NCS + k] = ncs;

    // n = sum(new_cluster_size): wave reduce then serial LDS combine
    float s = ncs;
    #pragma unroll
    for (int off = 16; off >= 1; off >>= 1) s += __shfl_xor(s, off, 32);
    if ((k & 31) == 0) red[k >> 5] = s;
    __syncthreads();
    float n = 0.0f;
    #pragma unroll
    for (int i = 0; i < 32; ++i) n += red[i];

    float cs  = (ncs + EPS) / (n + NUM_EMB * EPS) * n;
    float inv = 1.0f / cs;
    #pragma unroll
    for (int d = 0; d < EMB_DIM; ++d) {
        float nw = ema_w[k * EMB_DIM + d] * DECAY + OMD * ws[WS_DW + k * EMB_DIM + d];
        out[OUT_NEMAW  + k * EMB_DIM + d] = nw;
        out[OUT_NEWEMB + k * EMB_DIM + d] = nw * inv;
    }
    if (k == 0) out[OUT_LOSS] = COMMIT * ws[WS_LOSS] * (1.0f / 2097152.0f);
}

// ---------------------------------------------------------------------------
extern "C" void kernel_launch(void* const* d_in, const int* in_sizes, int n_in,
                              void* d_out, int out_size, void* d_ws, size_t ws_size,
                              hipStream_t stream) {
    const float* in    = (const float*)d_in[0];   // inputs  [32,64,32,32]
    const float* emb   = (const float*)d_in[1];   // embedding [1024,64]
    const float* emacs = (const float*)d_in[2];   // ema_cluster_size [1024]
    const float* emaw  = (const float*)d_in[3];   // ema_w [1024,64]
    float* out = (float*)d_out;
    float* ws  = (float*)d_ws;

    vq_init<<<(WS_TOTAL + 255) / 256, 256, 0, stream>>>(emb, ws);
    vq_main<<<N_FLAT / 16, 128, 0, stream>>>(in, emb, out, ws);
    vq_finalize<<<1, 1024, 0, stream>>>(emacs, emaw, out, ws);
}